// KPlaneHashGrid_32238024524551
// MI455X (gfx1250) — compile-verified
//
#include <hip/hip_runtime.h>
#include <stdint.h>

// K-Planes hash-grid encoding, MI455X (gfx1250, wave32).
// - One thread per (point, level): 8.4M threads, 12 independent 8B gathers each.
// - Block's 16 points (48 dwords) staged into LDS via global_load_async_to_lds_b32
//   (ASYNCcnt path, bypasses VGPRs) -> coords read from LDS with broadcast reads.
// - Tables (192 MiB == global L2 size) gathered with regular cached loads.
// - Output (64 MB stream) written with non-temporal stores to avoid evicting tables.

typedef float v2f __attribute__((ext_vector_type(2)));

#define NLVL   16
#define TSIZE  (1u << 19)
#define TMASK  ((1u << 19) - 1u)
#define PRIME1 2654435761u

// floor(16 * 1.3819^l) computed in f64 on host, hard-coded to match NumPy.
__constant__ float c_res[NLVL] = {
    16.f, 22.f, 30.f, 42.f, 58.f, 80.f, 111.f, 153.f,
    212.f, 294.f, 406.f, 561.f, 775.f, 1072.f, 1481.f, 2047.f
};

__global__ __launch_bounds__(256, 8)
void kplane_encode_kernel(const float* __restrict__ pts,
                          const float* __restrict__ tables,
                          float* __restrict__ out,
                          unsigned n_pts)
{
    // 16 points/block * 3 coords = 48 dwords of point data per block.
    __shared__ float s_pts[48];

    const unsigned tid = threadIdx.x;
    const unsigned t   = blockIdx.x * blockDim.x + tid;
    const unsigned pt  = t >> 4;      // point index
    const unsigned lvl = t & 15u;     // level index

    // ---- Async stage: pull this block's 48 coord dwords into LDS ----------
    // global_load_async_to_lds_b32 (GVS mode): mem = SADDR + VADDR32, dest LDS
    // byte address in a VGPR. Tracked by ASYNCcnt, data never touches VGPRs.
    {
        const unsigned elem = blockIdx.x * 48u + tid;   // dword index into pts
        if (tid < 48u && elem < n_pts * 3u) {
            const unsigned lds_addr = (unsigned)(uintptr_t)&s_pts[tid];
            const unsigned goff     = elem * 4u;        // byte offset from pts base
            asm volatile("global_load_async_to_lds_b32 %0, %1, %2"
                         :: "v"(lds_addr), "v"(goff), "s"(pts)
                         : "memory");
        }
        asm volatile("s_wait_asynccnt 0x0" ::: "memory");
        __syncthreads();
    }

    if (pt >= n_pts) return;

    // Broadcast LDS reads: 16 consecutive lanes share one point -> same address.
    const unsigned lp = (tid >> 4) * 3u;   // local point base in s_pts
    const float x = s_pts[lp + 0];
    const float y = s_pts[lp + 1];
    const float z = s_pts[lp + 2];
    const float res = c_res[lvl];

    // plane 0: (x,y)  plane 1: (x,z)  plane 2: (y,z)
    const float ca[3] = { x, x, y };
    const float cb[3] = { y, z, z };

    v2f   f[12];
    float w[12];

    // Phase 1: compute all 12 hash indices, issue all 12 gathers (max MLP).
#pragma unroll
    for (int p = 0; p < 3; ++p) {
        const float s0  = ca[p] * res;
        const float s1  = cb[p] * res;
        const float fl0 = floorf(s0);
        const float fl1 = floorf(s1);
        const float fr0 = s0 - fl0;
        const float fr1 = s1 - fl1;
        const unsigned u0 = (unsigned)fl0;
        const unsigned u1 = (unsigned)fl1;

        // hash = cx*PRIME0(=1) ^ cy*PRIME1 ; T = 2^19 -> modulo is a mask
        const unsigned h0 = u1 * PRIME1;
        const unsigned h1 = (u1 + 1u) * PRIME1;
        const unsigned i00 = ( u0        ^ h0) & TMASK;  // corner (0,0)
        const unsigned i01 = ( u0        ^ h1) & TMASK;  // corner (0,1)
        const unsigned i10 = ((u0 + 1u)  ^ h0) & TMASK;  // corner (1,0)
        const unsigned i11 = ((u0 + 1u)  ^ h1) & TMASK;  // corner (1,1)

        const v2f* __restrict__ tb =
            (const v2f*)tables + ((unsigned)p * NLVL + lvl) * TSIZE;

        f[4 * p + 0] = tb[i00];   // global_load_b64, cached (L2-resident tables)
        f[4 * p + 1] = tb[i01];
        f[4 * p + 2] = tb[i10];
        f[4 * p + 3] = tb[i11];

        const float g0 = 1.f - fr0;
        const float g1 = 1.f - fr1;
        w[4 * p + 0] = g0  * g1;
        w[4 * p + 1] = g0  * fr1;
        w[4 * p + 2] = fr0 * g1;
        w[4 * p + 3] = fr0 * fr1;
    }

    // Phase 2: bilinear reduce per plane (corner order 00,01,10,11), then
    // multiply the three planes elementwise.
    float o0 = 1.f, o1 = 1.f;
#pragma unroll
    for (int p = 0; p < 3; ++p) {
        float s0 = w[4 * p + 0] * f[4 * p + 0].x;
        float s1 = w[4 * p + 0] * f[4 * p + 0].y;
        s0 = fmaf(w[4 * p + 1], f[4 * p + 1].x, s0);
        s1 = fmaf(w[4 * p + 1], f[4 * p + 1].y, s1);
        s0 = fmaf(w[4 * p + 2], f[4 * p + 2].x, s0);
        s1 = fmaf(w[4 * p + 2], f[4 * p + 2].y, s1);
        s0 = fmaf(w[4 * p + 3], f[4 * p + 3].x, s0);
        s1 = fmaf(w[4 * p + 3], f[4 * p + 3].y, s1);
        o0 *= s0;
        o1 *= s1;
    }

    // out[pt*32 + lvl*2 + {0,1}] == ((v2f*)out)[t] -> wave writes 256B contiguous.
    // Non-temporal: don't let the output stream evict L2-resident tables.
    v2f o;
    o.x = o0;
    o.y = o1;
    __builtin_nontemporal_store(o, (v2f*)out + t);
}

extern "C" void kernel_launch(void* const* d_in, const int* in_sizes, int n_in,
                              void* d_out, int out_size, void* d_ws, size_t ws_size,
                              hipStream_t stream)
{
    const float* pts    = (const float*)d_in[0];   // (N, 3) f32
    const float* tables = (const float*)d_in[1];   // (3, 16, 2^19, 2) f32
    float*       out    = (float*)d_out;           // (N, 32) f32

    const unsigned n_pts = (unsigned)(in_sizes[0] / 3);
    const unsigned total = n_pts * 16u;            // one thread per (pt, level)

    dim3 block(256);
    dim3 grid((total + block.x - 1) / block.x);
    hipLaunchKernelGGL(kplane_encode_kernel, grid, block, 0, stream,
                       pts, tables, out, n_pts);
}